// Model_13357348291073
// MI455X (gfx1250) — compile-verified
//
#include <hip/hip_runtime.h>

#define N_NODES   50000
#define E_EDGES   800000
#define IN_DIM    128
#define H_DIM     96
#define OUT_DIM   64
#define NUM_LAYERS 4

typedef __attribute__((ext_vector_type(2))) float v2f;
typedef __attribute__((ext_vector_type(8))) float v8f;

// D = A(16x4 f32) * B(4x16 f32) + C(16x16 f32)   -> v_wmma_f32_16x16x4_f32
__device__ __forceinline__ v8f wmma_f32_16x16x4(v2f a, v2f b, v8f c) {
  return __builtin_amdgcn_wmma_f32_16x16x4_f32(
      /*neg_a=*/false, a, /*neg_b=*/false, b,
      /*c_mod=*/(short)0, c, /*reuse_a=*/false, /*reuse_b=*/false);
}

// ---------------------------------------------------------------------------
// Repack B [KDIM x NDIM] row-major into WMMA fragment order:
//   P[((s*NT + t)*32 + lane)] = ( B[4s+koff][16t+col], B[4s+koff+1][16t+col] )
// with koff = (lane>>4)*2, col = lane&15.  One float2 per lane -> the GEMM
// reads B as a single conflict-free 8B load per tile per k-step.
// blockIdx.y selects a matrix (head) via strides.
// ---------------------------------------------------------------------------
__global__ void pack_b_kernel(const float* __restrict__ B0, long strideB,
                              float2* __restrict__ P0, long strideP,
                              int KDIM, int NDIM)
{
  const float* __restrict__ B = B0 + (long)blockIdx.y * strideB;
  float2* __restrict__ P      = P0 + (long)blockIdx.y * strideP;
  const int NT = NDIM / 16;
  const int total = (KDIM / 4) * NT * 32;
  int i = blockIdx.x * blockDim.x + threadIdx.x;
  if (i >= total) return;
  int lane = i & 31;
  int rest = i >> 5;
  int t = rest % NT;
  int s = rest / NT;
  int kr = 4 * s + ((lane >> 4) << 1);
  int c  = 16 * t + (lane & 15);
  float2 v;
  v.x = B[(long)kr * NDIM + c];
  v.y = B[(long)(kr + 1) * NDIM + c];
  P[i] = v;
}

// ---------------------------------------------------------------------------
// GEMM: one wave computes a 16-row strip of D (all NDIM columns).
// A: [M x KDIM] row-major.  Bp: packed fragments (see pack_b_kernel), staged
// in LDS once per block.  Inner loop: load ALL NT B-fragments into distinct
// registers (one DS clause, one wait), then burst the NT WMMAs.
// blockIdx.y selects a head: Bp += y*strideBp, D += y*strideD.
// ---------------------------------------------------------------------------
template <int KDIM, int NDIM, bool RELU>
__global__ void wmma_gemm_kernel(const float* __restrict__ A,
                                 const float2* __restrict__ Bp0, long strideBp,
                                 const float* __restrict__ bias,
                                 float* __restrict__ D0, long strideD,
                                 int Mtiles)
{
  constexpr int NT    = NDIM / 16;
  constexpr int STEPS = KDIM / 4;
  __shared__ float2 Blds[STEPS * NT * 32];   // == KDIM*NDIM*4 bytes (max 48KB)

  const float2* __restrict__ Bp = Bp0 + (long)blockIdx.y * strideBp;
  float* __restrict__ D         = D0  + (long)blockIdx.y * strideD;

  // Cooperative stage of packed B into LDS (fully coalesced b64 loads).
  for (int i = threadIdx.x; i < STEPS * NT * 32; i += blockDim.x)
    Blds[i] = Bp[i];
  __syncthreads();

  const int wave = blockIdx.x * (blockDim.x >> 5) + (threadIdx.x >> 5);
  if (wave >= Mtiles) return;            // wave-uniform: EXEC stays all-ones

  const int lane = threadIdx.x & 31;
  const int row  = lane & 15;            // A: M index for this lane
  const int koff = (lane >> 4) << 1;     // K sub-offset 0 or 2 per lane half
  const int col  = lane & 15;            // D: N index within tile

  const float* __restrict__ Arow = A + ((long)wave * 16 + row) * KDIM + koff;

  v8f acc[NT];
#pragma unroll
  for (int t = 0; t < NT; ++t) acc[t] = (v8f){0.f,0.f,0.f,0.f,0.f,0.f,0.f,0.f};

#pragma unroll 4
  for (int s = 0; s < STEPS; ++s) {
    v2f a = *(const v2f*)(Arow + 4 * s);             // global_load_b64
    const float2* bl = &Blds[(s * NT) * 32 + lane];

    // Phase 1: gather all B fragments for this k-step (single DS clause).
    v2f bfrag[NT];
#pragma unroll
    for (int t = 0; t < NT; ++t) {
      float2 bv = bl[t * 32];                        // ds_load_b64, bank-clean
      bfrag[t].x = bv.x;
      bfrag[t].y = bv.y;
    }
    // Phase 2: WMMA burst (independent accumulator chains co-execute).
#pragma unroll
    for (int t = 0; t < NT; ++t)
      acc[t] = wmma_f32_16x16x4(a, bfrag[t], acc[t]);
  }

  // C/D layout: VGPR i holds M=i (lanes 0-15) and M=i+8 (lanes 16-31)
  const int rsel = (lane >> 4) << 3;     // 0 or 8
#pragma unroll
  for (int t = 0; t < NT; ++t) {
    const float bb = bias ? bias[t * 16 + col] : 0.0f;
#pragma unroll
    for (int i = 0; i < 8; ++i) {
      float v = acc[t][i] + bb;
      if (RELU) v = fmaxf(v, 0.0f);
      D[((long)wave * 16 + rsel + i) * NDIM + t * 16 + col] = v;
    }
  }
}

// comb[e,:] = ew0[e]*m0[src[e],:] + ew1[e]*m1[src[e],:]; acc[dst[e],:] += comb
// 24 threads per edge, float4 per thread; fp32 HW atomics into L2-resident acc.
__global__ void edge_scatter_kernel(const int* __restrict__ src,
                                    const int* __restrict__ dst,
                                    const float* __restrict__ ew0,
                                    const float* __restrict__ ew1,
                                    const float* __restrict__ m0,
                                    const float* __restrict__ m1,
                                    float* __restrict__ acc)
{
  constexpr int CH = H_DIM / 4;  // 24 float4 chunks per edge
  long tid = (long)blockIdx.x * blockDim.x + threadIdx.x;
  int e = (int)(tid / CH);
  if (e >= E_EDGES) return;
  int q = (int)(tid % CH) * 4;

  int   s  = src[e];
  int   d  = dst[e];
  float w0 = ew0[e];
  float w1 = ew1[e];

  float4 a = *(const float4*)(m0 + (long)s * H_DIM + q);
  float4 b = *(const float4*)(m1 + (long)s * H_DIM + q);

  float* out = acc + (long)d * H_DIM + q;
  unsafeAtomicAdd(out + 0, w0 * a.x + w1 * b.x);
  unsafeAtomicAdd(out + 1, w0 * a.y + w1 * b.y);
  unsafeAtomicAdd(out + 2, w0 * a.z + w1 * b.z);
  unsafeAtomicAdd(out + 3, w0 * a.w + w1 * b.w);
}

__global__ void zero_kernel(float4* __restrict__ p, int n4) {
  int i = blockIdx.x * blockDim.x + threadIdx.x;
  if (i < n4) p[i] = make_float4(0.f, 0.f, 0.f, 0.f);
}

__global__ void relu_kernel(const float4* __restrict__ in, float4* __restrict__ out, int n4) {
  int i = blockIdx.x * blockDim.x + threadIdx.x;
  if (i < n4) {
    float4 v = in[i];
    out[i] = make_float4(fmaxf(v.x, 0.f), fmaxf(v.y, 0.f),
                         fmaxf(v.z, 0.f), fmaxf(v.w, 0.f));
  }
}

__global__ void copy_kernel(const float4* __restrict__ in, float4* __restrict__ out, int n4) {
  int i = blockIdx.x * blockDim.x + threadIdx.x;
  if (i < n4) out[i] = in[i];
}

extern "C" void kernel_launch(void* const* d_in, const int* in_sizes, int n_in,
                              void* d_out, int out_size, void* d_ws, size_t ws_size,
                              hipStream_t stream)
{
  const float* x       = (const float*)d_in[0];
  const int*   eidx    = (const int*)  d_in[1];
  const float* enc_w   = (const float*)d_in[2];
  const float* enc_b   = (const float*)d_in[3];
  const float* dec_w   = (const float*)d_in[4];
  const float* dec_b   = (const float*)d_in[5];
  const float* conv_W  = (const float*)d_in[6];
  const float* conv_ew = (const float*)d_in[7];
  float* out = (float*)d_out;

  const int* src = eidx;            // edge_index[0]
  const int* dst = eidx + E_EDGES;  // edge_index[1]

  // Workspace layout (fp32): h | m (2 heads) | acc | packed weights
  float* h   = (float*)d_ws;
  float* m   = h + (long)N_NODES * H_DIM;
  float* acc = m + 2L * N_NODES * H_DIM;

  const int ENC_PK  = (IN_DIM / 4) * (H_DIM / 16) * 32;   // 6144 float2
  const int CONV_PK = (H_DIM / 4) * (H_DIM / 16) * 32;    // 4608 float2 per head-mat
  const int DEC_PK  = (H_DIM / 4) * (OUT_DIM / 16) * 32;  // 3072 float2

  float2* penc  = (float2*)(acc + (long)N_NODES * H_DIM);
  float2* pconv = penc + ENC_PK;                           // 8 matrices
  float2* pdec  = pconv + (long)NUM_LAYERS * 2 * CONV_PK;

  const int Mtiles = N_NODES / 16;  // 3125 exactly
  dim3 blk(256);                    // 8 waves per block
  const int wpb = 256 / 32;
  dim3 g_gemm((Mtiles + wpb - 1) / wpb, 1);

  const int n4 = N_NODES * H_DIM / 4;
  dim3 g_elem((n4 + 255) / 256);
  long edge_threads = (long)E_EDGES * (H_DIM / 4);
  dim3 g_edge((unsigned)((edge_threads + 255) / 256));

  // --- Pack all weight matrices into WMMA fragment order (once per call) ---
  pack_b_kernel<<<dim3((ENC_PK + 255) / 256, 1), blk, 0, stream>>>(
      enc_w, 0L, penc, 0L, IN_DIM, H_DIM);
  pack_b_kernel<<<dim3((CONV_PK + 255) / 256, NUM_LAYERS * 2), blk, 0, stream>>>(
      conv_W, (long)H_DIM * H_DIM, pconv, (long)CONV_PK, H_DIM, H_DIM);
  pack_b_kernel<<<dim3((DEC_PK + 255) / 256, 1), blk, 0, stream>>>(
      dec_w, 0L, pdec, 0L, H_DIM, OUT_DIM);

  // --- Encoder: h = relu(x @ enc_w + enc_b) ---
  wmma_gemm_kernel<IN_DIM, H_DIM, true><<<g_gemm, blk, 0, stream>>>(
      x, penc, 0L, enc_b, h, 0L, Mtiles);

  for (int l = 0; l < NUM_LAYERS; ++l) {
    const float2* Wpl = pconv + (long)l * 2 * CONV_PK;
    const float*  ew0 = conv_ew + ((long)l * 2 + 0) * E_EDGES;
    const float*  ew1 = conv_ew + ((long)l * 2 + 1) * E_EDGES;

    // m[head] = h @ W[l,head]  (grid.y = heads)
    dim3 g2(g_gemm.x, 2);
    wmma_gemm_kernel<H_DIM, H_DIM, false><<<g2, blk, 0, stream>>>(
        h, Wpl, (long)CONV_PK, nullptr, m, (long)N_NODES * H_DIM, Mtiles);

    zero_kernel<<<g_elem, blk, 0, stream>>>((float4*)acc, n4);

    edge_scatter_kernel<<<g_edge, blk, 0, stream>>>(
        src, dst, ew0, ew1, m, m + (long)N_NODES * H_DIM, acc);

    relu_kernel<<<g_elem, blk, 0, stream>>>((const float4*)acc, (float4*)h, n4);
  }

  // --- Decoder: out[:N*64] = h @ dec_w + dec_b ; out[N*64:] = h ---
  wmma_gemm_kernel<H_DIM, OUT_DIM, false><<<g_gemm, blk, 0, stream>>>(
      h, pdec, 0L, dec_b, out, 0L, Mtiles);
  copy_kernel<<<g_elem, blk, 0, stream>>>(
      (const float4*)h, (float4*)(out + (long)N_NODES * OUT_DIM), n4);
}